// GATv2Layer_25305947308809
// MI455X (gfx1250) — compile-verified
//
#include <hip/hip_runtime.h>
#include <hip/hip_bf16.h>

typedef __attribute__((ext_vector_type(2))) float v2f;
typedef __attribute__((ext_vector_type(8))) float v8f;

#define HF  4     // heads
#define FF  64    // feature dim per head (= in dim = edge dim)
#define HFD 256   // HF*FF

// ---- order-preserving float <-> uint encoding for atomic max ----
__device__ __forceinline__ unsigned fenc(float f) {
  unsigned u = __float_as_uint(f);
  return (u & 0x80000000u) ? ~u : (u | 0x80000000u);
}
__device__ __forceinline__ float fdec(unsigned u) {
  return (u & 0x80000000u) ? __uint_as_float(u ^ 0x80000000u)
                           : __uint_as_float(~u);
}

// ---- CDNA5 async global->LDS copy (16B per lane), ASYNCcnt-tracked ----
__device__ __forceinline__ void async_g2l_b128(unsigned lds_addr, unsigned long long gaddr) {
  asm volatile("global_load_async_to_lds_b128 %0, %1, off"
               :: "v"(lds_addr), "v"(gaddr) : "memory");
}
__device__ __forceinline__ void wait_async0() {
  asm volatile("s_wait_asynccnt 0x0" ::: "memory");
}

// ---------------- init ----------------
__global__ __launch_bounds__(256) void k_init_amax(unsigned* __restrict__ amax, long n) {
  long i = (long)blockIdx.x * 256 + threadIdx.x;
  if (i < n) amax[i] = 0x00800000u;  // fenc(-FLT_MAX)
}

// ---- pack a [64,256] weight into WMMA-B-tile order:
// P[(ct*16+kt)*64 + 2*l + j] = W[(4*kt + 2*(l>>4) + j)*256 + ct*16 + (l&15)]
__global__ __launch_bounds__(256) void k_pack_b(const float* __restrict__ W,
                                                float* __restrict__ P) {
  int t = blockIdx.x * 256 + threadIdx.x;     // 16*16*32 = 8192
  if (t >= 16 * 16 * 32) return;
  int l  = t & 31;
  int kt = (t >> 5) & 15;
  int ct = t >> 9;
  int k   = 4 * kt + 2 * (l >> 4);
  int col = ct * 16 + (l & 15);
  size_t o = (size_t)(ct * 16 + kt) * 64 + 2 * l;
  P[o + 0] = W[(size_t)k * HFD + col];
  P[o + 1] = W[(size_t)(k + 1) * HFD + col];
}

// ---------------- self-loop attr = mean of incoming edge_attr ----------------
__global__ __launch_bounds__(256) void k_loop_scatter(const float* __restrict__ ea,
    const int* __restrict__ dst, float* __restrict__ lsum, float* __restrict__ cnt, long E) {
  long t = (long)blockIdx.x * 256 + threadIdx.x;
  if (t >= E * FF) return;
  long e = t >> 6; int f = (int)(t & 63);
  int d = dst[e];
  atomicAdd(&lsum[(size_t)d * FF + f], ea[(size_t)e * FF + f]);
  if (f == 0) atomicAdd(&cnt[d], 1.0f);
}

__global__ __launch_bounds__(256) void k_loop_div(float* __restrict__ lsum,
    const float* __restrict__ cnt, long N) {
  long t = (long)blockIdx.x * 256 + threadIdx.x;
  if (t >= N * FF) return;
  float c = cnt[t >> 6];
  lsum[t] *= 1.0f / (c < 1.0f ? 1.0f : c);
}

// ---------------- XL = x@W_l + b_l, XR = x@W_r + b_r (f32 WMMA, packed B) ----------------
__global__ __launch_bounds__(32) void k_gemm_xlr(const float* __restrict__ x,
    const float* __restrict__ PWl, const float* __restrict__ bl,
    const float* __restrict__ PWr, const float* __restrict__ br,
    float* __restrict__ XL, float* __restrict__ XR, long N)
{
  const int lane = threadIdx.x;
  const int m  = lane & 15;
  const int kg = lane >> 4;
  const long row0 = (long)blockIdx.x * 16;

  long rr = row0 + m; if (rr >= N) rr = N - 1;
  const float* xp = x + (size_t)rr * FF;
  v2f a[16];
  #pragma unroll
  for (int kt = 0; kt < 16; ++kt) {
    int k = 4 * kt + 2 * kg;
    a[kt].x = xp[k]; a[kt].y = xp[k + 1];
  }

  const v2f* pl = (const v2f*)PWl;
  const v2f* pr = (const v2f*)PWr;

  #pragma unroll 1
  for (int ct = 0; ct < 16; ++ct) {
    v8f cl = {}; v8f cr = {};
    #pragma unroll
    for (int kt = 0; kt < 16; ++kt) {
      v2f bL = pl[(size_t)(ct * 16 + kt) * 32 + lane];
      cl = __builtin_amdgcn_wmma_f32_16x16x4_f32(false, a[kt], false, bL,
                                                 (short)0, cl, false, false);
      v2f bR = pr[(size_t)(ct * 16 + kt) * 32 + lane];
      cr = __builtin_amdgcn_wmma_f32_16x16x4_f32(false, a[kt], false, bR,
                                                 (short)0, cr, false, false);
    }
    int col = ct * 16 + m;
    float bvl = bl[col], bvr = br[col];
    #pragma unroll
    for (int r = 0; r < 8; ++r) {
      long row = row0 + r + 8 * kg;
      if (row < N) {
        XL[(size_t)row * HFD + col] = cl[r] + bvl;
        XR[(size_t)row * HFD + col] = cr[r] + bvr;
      }
    }
  }
}

// ---- fused: e_emb = ea@W_e ; z = XL[src]+XR[dst]+e_emb ; alpha = att . leaky(z) ----
__global__ __launch_bounds__(32) void k_edge_alpha(
    const float* __restrict__ edge_attr, const float* __restrict__ loop_attr,
    const int* __restrict__ eidx, const float* __restrict__ PWe,
    const float* __restrict__ att, const float* __restrict__ XL,
    const float* __restrict__ XR, float* __restrict__ alpha, long E, long TE)
{
  // stride 260 floats (1040 B): rows r and r+8 land 32 banks apart -> conflict-free
  __shared__ float s_xl[16][260];
  __shared__ float s_xr[16][260];
  const int lane = threadIdx.x;
  const int m  = lane & 15;
  const int kg = lane >> 4;
  const long e0 = (long)blockIdx.x * 16;

  // async-stage XL[src] and XR[dst] rows (memory -> LDS, no VGPR round trip)
  unsigned lxl = (unsigned)(size_t)&s_xl[0][0];
  unsigned lxr = (unsigned)(size_t)&s_xr[0][0];
  unsigned lo = (unsigned)lane * 16u;
  for (int i = 0; i < 16; ++i) {
    long e = e0 + i; if (e >= TE) e = TE - 1;
    int s, d;
    if (e < E) { s = eidx[e]; d = eidx[E + e]; } else { s = d = (int)(e - E); }
    unsigned long long gl = (unsigned long long)(size_t)(XL + (size_t)s * HFD);
    unsigned long long gr = (unsigned long long)(size_t)(XR + (size_t)d * HFD);
    unsigned ld = (unsigned)i * 1040u + lo;
    async_g2l_b128(lxl + ld,        gl + lo);
    async_g2l_b128(lxl + ld + 512u, gl + lo + 512u);
    async_g2l_b128(lxr + ld,        gr + lo);
    async_g2l_b128(lxr + ld + 512u, gr + lo + 512u);
  }

  // A-tile: 16 edges x 64 of ea (edge_attr or loop_attr), loaded while async runs
  const float* ap;
  {
    long e = e0 + m; if (e >= TE) e = TE - 1;
    ap = (e < E) ? (edge_attr + (size_t)e * FF) : (loop_attr + (size_t)(e - E) * FF);
  }
  v2f a[16];
  #pragma unroll
  for (int kt = 0; kt < 16; ++kt) {
    int k = 4 * kt + 2 * kg;
    a[kt].x = ap[k]; a[kt].y = ap[k + 1];
  }

  const v2f* pw = (const v2f*)PWe;

  wait_async0();
  __syncthreads();

  #pragma unroll 1
  for (int h = 0; h < HF; ++h) {
    float acc[8] = {0.f, 0.f, 0.f, 0.f, 0.f, 0.f, 0.f, 0.f};
    #pragma unroll 1
    for (int c4 = 0; c4 < 4; ++c4) {
      int ct = h * 4 + c4;
      v8f c = {};
      #pragma unroll
      for (int kt = 0; kt < 16; ++kt) {
        v2f bb = pw[(size_t)(ct * 16 + kt) * 32 + lane];
        c = __builtin_amdgcn_wmma_f32_16x16x4_f32(false, a[kt], false, bb,
                                                  (short)0, c, false, false);
      }
      float av = att[ct * 16 + m];   // att[h][(ct%4)*16 + n] flattened
      #pragma unroll
      for (int r = 0; r < 8; ++r) {
        int rm = r + 8 * kg, cn = ct * 16 + m;
        float z = c[r] + s_xl[rm][cn] + s_xr[rm][cn];
        z = (z > 0.f) ? z : 0.2f * z;
        acc[r] += av * z;
      }
    }
    // reduce each row's partial over the 16 lanes of each half-wave
    #pragma unroll
    for (int r = 0; r < 8; ++r) {
      float v = acc[r];
      v += __shfl_xor(v, 1, 32);
      v += __shfl_xor(v, 2, 32);
      v += __shfl_xor(v, 4, 32);
      v += __shfl_xor(v, 8, 32);
      if (m == 0) {
        long e = e0 + r + 8 * kg;
        if (e < TE) alpha[e * HF + h] = v;
      }
    }
  }
}

// ---------------- segment max / exp-sum / weighted scatter ----------------
__global__ __launch_bounds__(256) void k_amax(const float* __restrict__ alpha,
    const int* __restrict__ eidx, unsigned* __restrict__ amax, long E, long TE) {
  long t = (long)blockIdx.x * 256 + threadIdx.x;
  if (t >= TE * HF) return;
  long e = t >> 2;
  int d = (e < E) ? eidx[E + e] : (int)(e - E);
  atomicMax(&amax[(size_t)d * HF + (t & 3)], fenc(alpha[t]));
}

__global__ __launch_bounds__(256) void k_expsum(float* __restrict__ alpha,
    const int* __restrict__ eidx, const unsigned* __restrict__ amax,
    float* __restrict__ denom, long E, long TE) {
  long t = (long)blockIdx.x * 256 + threadIdx.x;
  if (t >= TE * HF) return;
  long e = t >> 2; int h = (int)(t & 3);
  int d = (e < E) ? eidx[E + e] : (int)(e - E);
  float ex = expf(alpha[t] - fdec(amax[(size_t)d * HF + h]));
  alpha[t] = ex;
  atomicAdd(&denom[(size_t)d * HF + h], ex);
}

__global__ __launch_bounds__(256) void k_scatter_out(const float* __restrict__ alpha,
    const int* __restrict__ eidx, const float* __restrict__ denom,
    const float* __restrict__ XL, float* __restrict__ outp, long E, long TE) {
  long t = (long)blockIdx.x * 256 + threadIdx.x;
  if (t >= TE * FF) return;
  long e = t >> 6; int f = (int)(t & 63);
  int s, d;
  if (e < E) { s = eidx[e]; d = eidx[E + e]; } else { s = d = (int)(e - E); }
  float acc = 0.f;
  #pragma unroll
  for (int h = 0; h < HF; ++h) {
    float al = alpha[e * HF + h] / (denom[(size_t)d * HF + h] + 1e-16f);
    acc += al * XL[(size_t)s * HFD + h * FF + f];
  }
  atomicAdd(&outp[(size_t)d * FF + f], 0.25f * acc);  // /H folded in
}

// ---------------- batch norm (training stats) + relu ----------------
__global__ __launch_bounds__(256) void k_bn_stats(const float* __restrict__ outp,
    const float* __restrict__ bias, float* __restrict__ sums,
    float* __restrict__ sumsq, long N) {
  __shared__ float s1[256], s2[256];
  int f  = threadIdx.x & 63;
  int rg = threadIdx.x >> 6;       // 0..3
  float a1 = 0.f, a2 = 0.f;
  float bv = bias[f];
  for (long r = (long)blockIdx.x * 4 + rg; r < N; r += (long)gridDim.x * 4) {
    float v = outp[r * FF + f] + bv;
    a1 += v; a2 += v * v;
  }
  s1[threadIdx.x] = a1; s2[threadIdx.x] = a2;
  __syncthreads();
  if (threadIdx.x < 64) {
    float t1 = s1[threadIdx.x] + s1[threadIdx.x + 64] + s1[threadIdx.x + 128] + s1[threadIdx.x + 192];
    float t2 = s2[threadIdx.x] + s2[threadIdx.x + 64] + s2[threadIdx.x + 128] + s2[threadIdx.x + 192];
    atomicAdd(&sums[threadIdx.x],  t1);
    atomicAdd(&sumsq[threadIdx.x], t2);
  }
}

__global__ __launch_bounds__(256) void k_bn_apply(const float* __restrict__ outp,
    const float* __restrict__ bias, const float* __restrict__ gamma,
    const float* __restrict__ beta, const float* __restrict__ sums,
    const float* __restrict__ sumsq, float* __restrict__ out, long N) {
  long t = (long)blockIdx.x * 256 + threadIdx.x;
  if (t >= N * FF) return;
  int f = (int)(t & 63);
  float invN = 1.0f / (float)N;
  float mu  = sums[f] * invN;
  float var = sumsq[f] * invN - mu * mu;
  float v = outp[t] + bias[f];
  float o = gamma[f] * (v - mu) * rsqrtf(var + 1e-5f) + beta[f];
  out[t] = o > 0.f ? o : 0.f;
}

// ---------------- launch ----------------
extern "C" void kernel_launch(void* const* d_in, const int* in_sizes, int n_in,
                              void* d_out, int out_size, void* d_ws, size_t ws_size,
                              hipStream_t stream) {
  const float* x    = (const float*)d_in[0];
  const int*   eidx = (const int*)  d_in[1];
  const float* ea   = (const float*)d_in[2];
  const float* Wl   = (const float*)d_in[3];
  const float* bl   = (const float*)d_in[4];
  const float* Wr   = (const float*)d_in[5];
  const float* br   = (const float*)d_in[6];
  const float* We   = (const float*)d_in[7];
  const float* att  = (const float*)d_in[8];
  const float* bias = (const float*)d_in[9];
  const float* gamma= (const float*)d_in[10];
  const float* beta = (const float*)d_in[11];
  float* out = (float*)d_out;

  const long N  = in_sizes[0] / FF;
  const long E  = in_sizes[1] / 2;
  const long TE = E + N;

  char* w = (char*)d_ws;
  size_t off = 0;
  auto take = [&](size_t nbytes) -> char* {
    char* p = w + off;
    off += (nbytes + 255) & ~(size_t)255;
    return p;
  };
  float*    XL    = (float*)   take((size_t)N * HFD * 4);
  float*    XR    = (float*)   take((size_t)N * HFD * 4);
  float*    loopA = (float*)   take((size_t)N * FF * 4);
  float*    cnt   = (float*)   take((size_t)N * 4);
  float*    alpha = (float*)   take((size_t)TE * HF * 4);
  unsigned* amax  = (unsigned*)take((size_t)N * HF * 4);
  float*    denom = (float*)   take((size_t)N * HF * 4);
  float*    outp  = (float*)   take((size_t)N * FF * 4);
  float*    sums  = (float*)   take(FF * 4);
  float*    sumsq = (float*)   take(FF * 4);
  float*    PWl   = (float*)   take((size_t)FF * HFD * 4);
  float*    PWr   = (float*)   take((size_t)FF * HFD * 4);
  float*    PWe   = (float*)   take((size_t)FF * HFD * 4);

  hipMemsetAsync(loopA, 0, (size_t)N * FF * 4, stream);
  hipMemsetAsync(cnt,   0, (size_t)N * 4,      stream);
  hipMemsetAsync(denom, 0, (size_t)N * HF * 4, stream);
  hipMemsetAsync(outp,  0, (size_t)N * FF * 4, stream);
  hipMemsetAsync(sums,  0, FF * 4, stream);
  hipMemsetAsync(sumsq, 0, FF * 4, stream);

  k_init_amax<<<(unsigned)((N * HF + 255) / 256), 256, 0, stream>>>(amax, N * HF);
  k_pack_b<<<32, 256, 0, stream>>>(Wl, PWl);
  k_pack_b<<<32, 256, 0, stream>>>(Wr, PWr);
  k_pack_b<<<32, 256, 0, stream>>>(We, PWe);
  k_loop_scatter<<<(unsigned)((E * FF + 255) / 256), 256, 0, stream>>>(ea, eidx + E, loopA, cnt, E);
  k_loop_div<<<(unsigned)((N * FF + 255) / 256), 256, 0, stream>>>(loopA, cnt, N);
  k_gemm_xlr<<<(unsigned)((N + 15) / 16), 32, 0, stream>>>(x, PWl, bl, PWr, br, XL, XR, N);
  k_edge_alpha<<<(unsigned)((TE + 15) / 16), 32, 0, stream>>>(ea, loopA, eidx, PWe, att, XL, XR, alpha, E, TE);
  k_amax<<<(unsigned)((TE * HF + 255) / 256), 256, 0, stream>>>(alpha, eidx, amax, E, TE);
  k_expsum<<<(unsigned)((TE * HF + 255) / 256), 256, 0, stream>>>(alpha, eidx, amax, denom, E, TE);
  k_scatter_out<<<(unsigned)((TE * FF + 255) / 256), 256, 0, stream>>>(alpha, eidx, denom, XL, outp, E, TE);
  k_bn_stats<<<512, 256, 0, stream>>>(outp, bias, sums, sumsq, N);
  k_bn_apply<<<(unsigned)((N * FF + 255) / 256), 256, 0, stream>>>(outp, bias, gamma, beta, sums, sumsq, out, N);
}